// EclDistanceMat_12335146074113
// MI455X (gfx1250) — compile-verified
//
#include <hip/hip_runtime.h>

typedef float v2f __attribute__((ext_vector_type(2)));
typedef float v4f __attribute__((ext_vector_type(4)));
typedef float v8f __attribute__((ext_vector_type(8)));

constexpr int BATCH = 2;
constexpr int N     = 2048;
constexpr int D     = 32;
constexpr int TILE  = 128;          // block tile edge (output)
constexpr float EPS = 1e-7f;

// Each block: 128x128 output tile of one batch.
// 8 waves; each wave computes 8 independent 16x16 WMMA tiles (one row of
// sub-tiles -> A fragment reuse). K=32 chained as 8x V_WMMA_F32_16X16X4_F32.
__launch_bounds__(256, 2)
__global__ void ecl_dist_wmma_kernel(const float* __restrict__ x,
                                     const float* __restrict__ y,
                                     float* __restrict__ out) {
    __shared__ float xs[TILE][D];     // 16 KB
    __shared__ float ys[TILE][D];     // 16 KB
    __shared__ float nxs[TILE];       // ||x_row||^2
    __shared__ float nys[TILE];       // ||y_row||^2

    const int b   = blockIdx.z;
    const int i0  = blockIdx.y * TILE;   // x-row base (output row)
    const int j0  = blockIdx.x * TILE;   // y-row base (output col)
    const int tid = threadIdx.x;         // 0..255

    // ---- Phase 1: stage tiles into LDS (float4, fully coalesced) ----
    {
        const v4f* xg4 = (const v4f*)(x + ((size_t)b * N + i0) * D);
        const v4f* yg4 = (const v4f*)(y + ((size_t)b * N + j0) * D);
        v4f* xs4 = (v4f*)&xs[0][0];
        v4f* ys4 = (v4f*)&ys[0][0];
        #pragma unroll
        for (int f = tid; f < TILE * D / 4; f += 256) {
            xs4[f] = xg4[f];
            ys4[f] = yg4[f];
        }
    }
    __syncthreads();

    // ---- Row norms: threads 0..127 do x rows, 128..255 do y rows ----
    if (tid < TILE) {
        float s = 0.f;
        #pragma unroll
        for (int k = 0; k < D; ++k) { float v = xs[tid][k]; s = fmaf(v, v, s); }
        nxs[tid] = s;
    } else {
        const int r = tid - TILE;
        float s = 0.f;
        #pragma unroll
        for (int k = 0; k < D; ++k) { float v = ys[r][k]; s = fmaf(v, v, s); }
        nys[r] = s;
    }
    __syncthreads();

    // ---- Phase 2: WMMA over 16x16 sub-tiles ----
    const int wave = tid >> 5;          // 0..7
    const int lane = tid & 31;
    const int half = lane >> 4;         // 0 or 1
    const int l16  = lane & 15;
    const int koff = half * 2;          // f32 A/B frag: lanes16-31 hold K+2,K+3

    const int tm   = wave;              // sub-tile row of this wave (0..7)
    const int mrow = tm * 16 + l16;     // x row inside LDS tile for A frag

    #pragma unroll
    for (int tn = 0; tn < 8; ++tn) {    // sub-tile col (0..7)
        const int nrow = tn * 16 + l16; // y row inside LDS tile for B frag

        v8f acc = {};                   // C = 0
        #pragma unroll
        for (int kk = 0; kk < D; kk += 4) {
            v2f a = *(const v2f*)&xs[mrow][kk + koff];   // ds_load_b64
            v2f bb = *(const v2f*)&ys[nrow][kk + koff];  // ds_load_b64
            acc = __builtin_amdgcn_wmma_f32_16x16x4_f32(
                /*neg_a=*/false, a, /*neg_b=*/false, bb,
                /*c_mod=*/(short)0, acc, /*reuse_a=*/false, /*reuse_b=*/false);
        }

        // Epilogue: d = sqrt(max(nx + ny - 2*dot, eps))
        const float nyv  = nys[tn * 16 + l16];
        const float* nxp = &nxs[tm * 16 + half * 8];     // 8 contiguous floats
        float* op = out + (((size_t)b * N + (size_t)(i0 + tm * 16 + half * 8)) * N
                           + (size_t)(j0 + tn * 16 + l16));
        #pragma unroll
        for (int r = 0; r < 8; ++r) {
            float d2 = fmaf(-2.0f, acc[r], nxp[r] + nyv);
            d2 = fmaxf(d2, EPS);
            __builtin_nontemporal_store(__builtin_sqrtf(d2), op + (size_t)r * N);
        }
    }
}

extern "C" void kernel_launch(void* const* d_in, const int* in_sizes, int n_in,
                              void* d_out, int out_size, void* d_ws, size_t ws_size,
                              hipStream_t stream) {
    const float* x = (const float*)d_in[0];
    const float* y = (const float*)d_in[1];
    float* out = (float*)d_out;

    dim3 grid(N / TILE, N / TILE, BATCH);   // (16, 16, 2)
    dim3 block(256);
    ecl_dist_wmma_kernel<<<grid, block, 0, stream>>>(x, y, out);
}